// Encoder_46042049413177
// MI455X (gfx1250) — compile-verified
//
#include <hip/hip_runtime.h>

typedef float v2f __attribute__((ext_vector_type(2)));
typedef float v8f __attribute__((ext_vector_type(8)));
typedef unsigned int uint32x4 __attribute__((ext_vector_type(4)));
typedef int int32x4 __attribute__((ext_vector_type(4)));
typedef int int32x8 __attribute__((ext_vector_type(8)));

#define TPB 256
#define WPB (TPB / 32)

static inline int ceil_div(int a, int b) { return (a + b - 1) / b; }

// ---------------- degree / symmetric norm ----------------
__global__ void k_deg_init(float* __restrict__ deg, int n) {
  int i = blockIdx.x * blockDim.x + threadIdx.x;
  if (i < n) deg[i] = 1.0f;  // self-loop contributes 1
}

__global__ void k_deg_accum(float* __restrict__ deg, const int* __restrict__ dst, int e) {
  int i = blockIdx.x * blockDim.x + threadIdx.x;
  if (i < e) atomicAdd(&deg[dst[i]], 1.0f);
}

__global__ void k_deg_rsqrt(float* __restrict__ deg, int n) {
  int i = blockIdx.x * blockDim.x + threadIdx.x;
  if (i < n) deg[i] = rsqrtf(deg[i]);  // deg -> d^{-1/2}
}

// ---------------- TDM: DMA a row-major [rows x cols] f32 panel into LDS ----------------
// D# descriptor built per CDNA5 ISA 8.3-8.6:
//   group0: count=1 | lds_addr | global_addr[56:0] | type=2
//   group1: data_size=4B, tensor_dim0=cols, tensor_dim1=rows, tile_dim0=cols,
//           tile_dim1=rows, tensor_dim0_stride=cols  (2D whole-panel copy)
__device__ static inline void tdm_load_panel(const float* gsrc, float* ldsDst,
                                             int rows, int cols) {
  unsigned int ldsOff = (unsigned int)(uintptr_t)ldsDst;  // addr[31:0] == LDS byte offset
  unsigned long long ga = (unsigned long long)(uintptr_t)gsrc;
  unsigned int td0 = (unsigned int)cols;
  unsigned int td1 = (unsigned int)rows;

  uint32x4 g0;
  g0.x = 1u;                                               // count=1, user mode
  g0.y = ldsOff;                                           // lds_addr
  g0.z = (unsigned int)ga;                                 // global_addr[31:0]
  g0.w = (unsigned int)((ga >> 32) & 0x01FFFFFFu) | (2u << 30);  // ga[56:32] | type=2

  int32x8 g1;
  g1[0] = (int)(2u << 16);                                 // data_size=2 (4 bytes), mask=0
  g1[1] = (int)((td0 & 0xFFFFu) << 16);                    // tensor_dim0[15:0] @ bits 63:48
  g1[2] = (int)(((td0 >> 16) & 0xFFFFu) | ((td1 & 0xFFFFu) << 16));  // td0 hi | td1 lo
  g1[3] = (int)(((td1 >> 16) & 0xFFFFu) | ((td0 & 0xFFFFu) << 16));  // td1 hi | tile_dim0
  g1[4] = (int)(td1 & 0xFFFFu);                            // tile_dim1=rows, tile_dim2=0
  g1[5] = (int)td0;                                        // tensor_dim0_stride low32
  g1[6] = 0;                                               // stride0 hi | stride1 lo
  g1[7] = 0;                                               // stride1 hi

  int32x4 gz = {0, 0, 0, 0};
#if __has_include(<hip/amd_detail/amd_gfx1250_TDM.h>)
  int32x8 gz8 = {0, 0, 0, 0, 0, 0, 0, 0};
  __builtin_amdgcn_tensor_load_to_lds(g0, g1, gz, gz, gz8, 0);   // 6-arg toolchain
#else
  __builtin_amdgcn_tensor_load_to_lds(g0, g1, gz, gz, 0);        // 5-arg (ROCm 7.2)
#endif
}

// ---------------- fp32 WMMA GEMM: C[M,Nc] = A[M,K] @ B[K,Nc] (+bias) ----------------
// B panel staged in LDS via TDM (issued by wave 0, s_wait_tensorcnt, block barrier).
// One wave owns a 16-row stripe of A and ALL TN column tiles (Nc == 16*TN):
// A fragment reused across column tiles; all TN B fragments loaded from LDS before
// the TN back-to-back WMMAs so a single wait covers the bundle.
// Layouts per CDNA5 ISA 7.12.2:
//   A 16x4 f32 : lanes 0-15 -> K={k0,k0+1}, lanes 16-31 -> K={k0+2,k0+3}, M = lane&15
//   B 4x16 f32 : rows striped across lanes; laneHi selects K half
//   C 16x16    : vgpr i -> M = i + 8*laneHi, N = lane&15
template <int TN>
__global__ void k_gemm_wmma(const float* __restrict__ A, const float* __restrict__ B,
                            const float* __restrict__ bias, float* __restrict__ C,
                            int M, int K, int Nc) {
  extern __shared__ float sB[];  // K * Nc floats

  if (threadIdx.x < 32) {        // wave-uniform: wave 0 issues the DMA
    tdm_load_panel(B, sB, K, Nc);
    __builtin_amdgcn_s_wait_tensorcnt(0);
  }
  __syncthreads();

  int wave = blockIdx.x * WPB + (threadIdx.x >> 5);
  int tilesM = M >> 4;
  if (wave >= tilesM) return;  // after barrier; wave-uniform, EXEC all-ones inside

  int lane   = threadIdx.x & 31;
  int laneLo = lane & 15;
  int laneHi = lane >> 4;

  v8f zero = {};
  v8f acc[TN];
#pragma unroll
  for (int t = 0; t < TN; ++t) acc[t] = zero;

  const float* aPtr = A + (size_t)((wave << 4) + laneLo) * K + (laneHi << 1);

  for (int k = 0; k < K; k += 4) {
    v2f a;
    a.x = aPtr[0];
    a.y = aPtr[1];
    int rb = (k + (laneHi << 1)) * Nc + laneLo;
    v2f bf[TN];
#pragma unroll
    for (int t = 0; t < TN; ++t) {   // batch all LDS reads first
      bf[t].x = sB[rb + t * 16];
      bf[t].y = sB[rb + Nc + t * 16];
    }
#pragma unroll
    for (int t = 0; t < TN; ++t)     // then TN independent WMMAs back-to-back
      acc[t] = __builtin_amdgcn_wmma_f32_16x16x4_f32(
          /*neg_a=*/false, a, /*neg_b=*/false, bf[t],
          /*c_mod=*/(short)0, acc[t], /*reuse_a=*/false, /*reuse_b=*/false);
    aPtr += 4;
  }

  int rbase = (wave << 4) + (laneHi << 3);
#pragma unroll
  for (int t = 0; t < TN; ++t) {
    int col  = t * 16 + laneLo;
    float bv = bias ? bias[col] : 0.0f;
#pragma unroll
    for (int i = 0; i < 8; ++i)
      C[(size_t)(rbase + i) * Nc + col] = acc[t][i] + bv;
  }
}

// ---------------- edge scatter: agg[dst] += norm * h[src] ----------------
// One wave per (edge or self-loop); lanes stride the feature dim -> coalesced
// gather + coalesced global_atomic_add_f32 stream (L2-resident node features).
__global__ void k_scatter(const float* __restrict__ h, float* __restrict__ agg,
                          const int* __restrict__ src, const int* __restrict__ dst,
                          const float* __restrict__ dis, int E, int N, int H) {
  int wave = blockIdx.x * WPB + (threadIdx.x >> 5);
  if (wave >= E + N) return;  // wave-uniform
  int lane = threadIdx.x & 31;

  int s, d;
  float norm;
  if (wave < E) {
    s = src[wave];
    d = dst[wave];
    norm = dis[s] * dis[d];
  } else {
    s = d = wave - E;  // self loop
    float t = dis[s];
    norm = t * t;
  }
  const float* hp = h + (size_t)s * H;
  float* ap = agg + (size_t)d * H;
  for (int f = lane; f < H; f += 32) atomicAdd(ap + f, hp[f] * norm);
}

// ---------------- bias + relu, in place ----------------
__global__ void k_bias_relu(float* __restrict__ a, const float* __restrict__ b, int total, int H) {
  int i = blockIdx.x * blockDim.x + threadIdx.x;
  if (i >= total) return;
  float v = a[i] + b[i % H];
  a[i] = v > 0.0f ? v : 0.0f;
}

// ---------------- global mean pool ----------------
__global__ void k_count(float* __restrict__ counts, const int* __restrict__ batch, int N) {
  int i = blockIdx.x * blockDim.x + threadIdx.x;
  if (i < N) atomicAdd(&counts[batch[i]], 1.0f);
}

__global__ void k_pool_sum(const float* __restrict__ h, float* __restrict__ pooled,
                           const int* __restrict__ batch, int N, int H) {
  int wave = blockIdx.x * WPB + (threadIdx.x >> 5);
  if (wave >= N) return;
  int lane = threadIdx.x & 31;
  int g = batch[wave];
  const float* hp = h + (size_t)wave * H;
  float* pp = pooled + (size_t)g * H;
  for (int f = lane; f < H; f += 32) atomicAdd(pp + f, hp[f]);
}

__global__ void k_pool_div(float* __restrict__ pooled, const float* __restrict__ counts,
                           int total, int H) {
  int i = blockIdx.x * blockDim.x + threadIdx.x;
  if (i >= total) return;
  pooled[i] /= fmaxf(counts[i / H], 1.0f);
}

// ---------------- launcher ----------------
extern "C" void kernel_launch(void* const* d_in, const int* in_sizes, int n_in,
                              void* d_out, int out_size, void* d_ws, size_t ws_size,
                              hipStream_t stream) {
  const float* x    = (const float*)d_in[0];
  const float* W1   = (const float*)d_in[1];
  const float* b1   = (const float*)d_in[2];
  const float* W2   = (const float*)d_in[3];
  const float* b2   = (const float*)d_in[4];
  const float* Wmu  = (const float*)d_in[5];
  const float* bmu  = (const float*)d_in[6];
  const int*   eidx = (const int*)d_in[7];
  const int*   batch = (const int*)d_in[8];

  const int N    = in_sizes[8];        // 100000
  const int E    = in_sizes[7] / 2;    // 1600000
  const int F_IN = in_sizes[0] / N;    // 128
  const int H1   = in_sizes[2];        // 160
  const int H2   = in_sizes[4];        // 80
  const int Z    = in_sizes[6];        // 64
  const int G    = out_size / Z;       // 512

  const int* src = eidx;
  const int* dst = eidx + E;

  // workspace carve-out (256B aligned): dis | hbuf (N*H1, reused for h2) | aggbuf (N*H1, reused) | pooled | counts
  char* ws = (char*)d_ws;
  auto pad = [](size_t b) { return (b + 255) & ~(size_t)255; };
  size_t offDis  = 0;
  size_t offH    = offDis + pad((size_t)N * sizeof(float));
  size_t offAgg  = offH + pad((size_t)N * H1 * sizeof(float));
  size_t offPool = offAgg + pad((size_t)N * H1 * sizeof(float));
  size_t offCnt  = offPool + pad((size_t)G * H2 * sizeof(float));
  float* dis    = (float*)(ws + offDis);
  float* hbuf   = (float*)(ws + offH);
  float* aggbuf = (float*)(ws + offAgg);
  float* pooled = (float*)(ws + offPool);
  float* counts = (float*)(ws + offCnt);

  // 1) degrees -> d^{-1/2}
  k_deg_init<<<ceil_div(N, TPB), TPB, 0, stream>>>(dis, N);
  k_deg_accum<<<ceil_div(E, TPB), TPB, 0, stream>>>(dis, dst, E);
  k_deg_rsqrt<<<ceil_div(N, TPB), TPB, 0, stream>>>(dis, N);

  const int tilesM = N / 16;  // 6250

  // 2) h1 = x @ W1   (TN = H1/16 = 10); B panel (80 KB) via TDM into LDS
  k_gemm_wmma<10><<<ceil_div(tilesM, WPB), TPB, (size_t)F_IN * H1 * sizeof(float), stream>>>(
      x, W1, nullptr, hbuf, N, F_IN, H1);

  // 3) agg1 = scatter(norm * h1) ; relu(agg1 + b1)
  hipMemsetAsync(aggbuf, 0, (size_t)N * H1 * sizeof(float), stream);
  k_scatter<<<ceil_div(E + N, WPB), TPB, 0, stream>>>(hbuf, aggbuf, src, dst, dis, E, N, H1);
  k_bias_relu<<<ceil_div(N * H1, TPB), TPB, 0, stream>>>(aggbuf, b1, N * H1, H1);

  // 4) h2 = agg1 @ W2   (TN = H2/16 = 5); B panel (50 KB) via TDM
  k_gemm_wmma<5><<<ceil_div(tilesM, WPB), TPB, (size_t)H1 * H2 * sizeof(float), stream>>>(
      aggbuf, W2, nullptr, hbuf, N, H1, H2);

  // 5) agg2 = scatter(norm * h2) ; relu(agg2 + b2), reuse aggbuf
  hipMemsetAsync(aggbuf, 0, (size_t)N * H2 * sizeof(float), stream);
  k_scatter<<<ceil_div(E + N, WPB), TPB, 0, stream>>>(hbuf, aggbuf, src, dst, dis, E, N, H2);
  k_bias_relu<<<ceil_div(N * H2, TPB), TPB, 0, stream>>>(aggbuf, b2, N * H2, H2);

  // 6) global mean pool
  hipMemsetAsync(pooled, 0, (size_t)G * H2 * sizeof(float), stream);
  hipMemsetAsync(counts, 0, (size_t)G * sizeof(float), stream);
  k_count<<<ceil_div(N, TPB), TPB, 0, stream>>>(counts, batch, N);
  k_pool_sum<<<ceil_div(N, WPB), TPB, 0, stream>>>(aggbuf, pooled, batch, N, H2);
  k_pool_div<<<ceil_div(G * H2, TPB), TPB, 0, stream>>>(pooled, counts, G * H2, H2);

  // 7) out = pooled @ Wmu + bmu   (TN = Z/16 = 4); B panel (20 KB) via TDM
  k_gemm_wmma<4><<<ceil_div(G / 16, WPB), TPB, (size_t)H2 * Z * sizeof(float), stream>>>(
      pooled, Wmu, bmu, (float*)d_out, G, H2, Z);
}